// GroupAwarePredictor_64424509440064
// MI455X (gfx1250) — compile-verified
//
#include <hip/hip_runtime.h>
#include <math.h>

#define N_PED 768
#define T_OBS 8
#define T_PRED 12
#define C_IN 2
#define C_OUT 5
#define HID 16
#define CG_IN 32          // (C_IN+2)*T_OBS
#define TH_ 1.0f
#define TAU_ 0.1f

typedef float v2f __attribute__((ext_vector_type(2)));
typedef float v8f __attribute__((ext_vector_type(8)));

// ---------------------------------------------------------------------------
// K1: build X[32][768] = concat(v_abs, velocity) with row cc = c*T_OBS + t
// ---------------------------------------------------------------------------
__global__ void k_build_X(const float* __restrict__ v_abs,
                          const float* __restrict__ v_rel,
                          const float* __restrict__ vel_w,
                          const float* __restrict__ vel_b,
                          float* __restrict__ X) {
  int idx = blockIdx.x * blockDim.x + threadIdx.x;
  if (idx >= CG_IN * N_PED) return;
  int cc = idx / N_PED, n = idx % N_PED;
  float val;
  if (cc < C_IN * T_OBS) {
    val = v_abs[cc * N_PED + n];
  } else {
    int o = (cc - C_IN * T_OBS) / T_OBS;
    int t = (cc - C_IN * T_OBS) % T_OBS;
    val = vel_w[o * C_IN + 0] * v_rel[(0 * T_OBS + t) * N_PED + n]
        + vel_w[o * C_IN + 1] * v_rel[(1 * T_OBS + t) * N_PED + n]
        + vel_b[o];
  }
  X[idx] = val;
}

// ---------------------------------------------------------------------------
// K2: Y[16,768] = W1[16,32] @ X[32,768] via V_WMMA_F32_16X16X4_F32
// one wave per 16-column tile; 8 WMMA steps over K=32
// ---------------------------------------------------------------------------
__global__ void __launch_bounds__(32) k_y_wmma(const float* __restrict__ W1,
                                               const float* __restrict__ X,
                                               float* __restrict__ Y) {
  int lane = threadIdx.x;
  int n0 = blockIdx.x * 16;
  int m = lane & 15;
  int hi = lane >> 4;
  v8f acc = {};
  for (int kk = 0; kk < CG_IN; kk += 4) {
    int ka = kk + 2 * hi;
    v2f a, b;
    a.x = W1[m * CG_IN + ka];
    a.y = W1[m * CG_IN + ka + 1];
    b.x = X[(ka + 0) * N_PED + n0 + m];
    b.y = X[(ka + 1) * N_PED + n0 + m];
    acc = __builtin_amdgcn_wmma_f32_16x16x4_f32(false, a, false, b,
                                                (short)0, acc, false, false);
  }
  for (int v = 0; v < 8; ++v) {
    int row = v + 8 * hi;
    Y[row * N_PED + n0 + m] = acc[v];
  }
}

// ---------------------------------------------------------------------------
// K3: dist_mean[i,j] = 0.5*(exp(s(i,j)) + exp(s(j,i))) fully fused
//     s(i,j) = sum_k w_k * relu(y_ki - y_kj + b1_k) + K0
// ---------------------------------------------------------------------------
__global__ void k_dist(const float* __restrict__ Y,
                       const float* __restrict__ g1_b,
                       const float* __restrict__ bn_g,
                       const float* __restrict__ bn_b,
                       const float* __restrict__ g2_w,
                       const float* __restrict__ g2_b,
                       float* __restrict__ DM) {
  int idx = blockIdx.x * blockDim.x + threadIdx.x;
  if (idx >= N_PED * N_PED) return;
  int i = idx / N_PED, j = idx % N_PED;
  const float inv_std = (float)(1.0 / sqrt(1.0 + 1e-5));
  float sij = 0.f, sji = 0.f, K0 = g2_b[0];
  for (int k = 0; k < HID; ++k) {
    float gk = g2_w[k];
    float w = gk * inv_std * bn_g[k];
    K0 += gk * bn_b[k];
    float dlt = Y[k * N_PED + i] - Y[k * N_PED + j];
    float b = g1_b[k];
    sij += w * fmaxf(dlt + b, 0.f);
    sji += w * fmaxf(-dlt + b, 0.f);
  }
  DM[idx] = 0.5f * (expf(sij + K0) + expf(sji + K0));
}

__device__ __forceinline__ float sig_of(float dm) {
  return 1.f / (1.f + expf((dm - TH_) / TAU_));
}

// ---------------------------------------------------------------------------
// K4: colsum[m] = sum_n sigmoid(-(DM[n,m]-TH)/TAU)
// ---------------------------------------------------------------------------
__global__ void k_colsum(const float* __restrict__ DM, float* __restrict__ colsum) {
  __shared__ float red[256];
  int m = blockIdx.x;
  float s = 0.f;
  for (int n = threadIdx.x; n < N_PED; n += 256)
    s += sig_of(DM[n * N_PED + m]);
  red[threadIdx.x] = s;
  __syncthreads();
  for (int off = 128; off > 0; off >>= 1) {
    if (threadIdx.x < off) red[threadIdx.x] += red[threadIdx.x + off];
    __syncthreads();
  }
  if (threadIdx.x == 0) colsum[m] = red[0];
}

// ---------------------------------------------------------------------------
// K5: VG[16,768]: v_soft = (Vrel @ sig)/colsum via WMMA, sigmoid fused into
//     B-load; epilogue vg = (vr - vsoft) + vsoft  (straight-through forward)
// ---------------------------------------------------------------------------
__global__ void __launch_bounds__(32) k_vsoft_wmma(const float* __restrict__ Vrel,
                                                   const float* __restrict__ DM,
                                                   const float* __restrict__ colsum,
                                                   float* __restrict__ VG) {
  int lane = threadIdx.x;
  int n0 = blockIdx.x * 16;
  int m = lane & 15;
  int hi = lane >> 4;
  v8f acc = {};
  for (int kk = 0; kk < N_PED; kk += 4) {
    int ka = kk + 2 * hi;
    v2f a, b;
    a.x = Vrel[m * N_PED + ka];
    a.y = Vrel[m * N_PED + ka + 1];
    b.x = sig_of(DM[(ka + 0) * N_PED + n0 + m]);
    b.y = sig_of(DM[(ka + 1) * N_PED + n0 + m]);
    acc = __builtin_amdgcn_wmma_f32_16x16x4_f32(false, a, false, b,
                                                (short)0, acc, false, false);
  }
  float cs = colsum[n0 + m];
  for (int v = 0; v < 8; ++v) {
    int row = v + 8 * hi;
    float vsoft = acc[v] / cs;
    float vr = Vrel[row * N_PED + n0 + m];
    VG[row * N_PED + n0 + m] = (vr - vsoft) + vsoft;
  }
}

// ---------------------------------------------------------------------------
// K6: faithful serial row-major relabeling + rank compaction (one block)
// ---------------------------------------------------------------------------
__global__ void __launch_bounds__(N_PED) k_group(const float* __restrict__ DM,
                                                 int* __restrict__ indices,
                                                 int* __restrict__ counts) {
  __shared__ int labels[N_PED];
  __shared__ unsigned masks[N_PED / 32];
  __shared__ int scan[N_PED];
  __shared__ int cnt[N_PED];
  int t = threadIdx.x;
  labels[t] = t;
  __syncthreads();

  for (int r = 1; r < N_PED; ++r) {
    if (t < N_PED / 32) masks[t] = 0u;
    __syncthreads();
    if (t < r && DM[r * N_PED + t] <= TH_)
      atomicOr(&masks[t >> 5], 1u << (t & 31));
    __syncthreads();
    for (int w = 0; w < N_PED / 32; ++w) {
      unsigned mw = masks[w];         // uniform across block
      while (mw) {
        int c = (w << 5) + (__ffs(mw) - 1);
        mw &= mw - 1;
        int L = labels[r];
        __syncthreads();
        if (labels[t] == L) labels[t] = c;
        __syncthreads();
      }
    }
    __syncthreads();
  }

  // rank surviving label values (sorted ascending) to 0..G-1
  scan[t] = 0;
  __syncthreads();
  scan[labels[t]] = 1;
  __syncthreads();
  for (int off = 1; off < N_PED; off <<= 1) {   // Hillis-Steele inclusive scan
    int v = scan[t];
    if (t >= off) v += scan[t - off];
    __syncthreads();
    scan[t] = v;
    __syncthreads();
  }
  int g = scan[labels[t]] - 1;
  indices[t] = g;
  cnt[t] = 0;
  __syncthreads();
  atomicAdd(&cnt[g], 1);
  __syncthreads();
  counts[t] = cnt[t];
}

// ---------------------------------------------------------------------------
// K7: pooling (zero + scatter-add)
// ---------------------------------------------------------------------------
__global__ void k_zero(float* __restrict__ p, int n) {
  int i = blockIdx.x * blockDim.x + threadIdx.x;
  if (i < n) p[i] = 0.f;
}

__global__ void k_scatter(const float* __restrict__ VG,
                          const int* __restrict__ indices,
                          float* __restrict__ pool) {
  int n = blockIdx.x * blockDim.x + threadIdx.x;
  if (n >= N_PED) return;
  int g = indices[n];
  for (int ct = 0; ct < C_IN * T_OBS; ++ct)
    atomicAdd(&pool[ct * N_PED + g], VG[ct * N_PED + n]);
}

// ---------------------------------------------------------------------------
// K8: out[o,p,n] = (baseline(v_rel) + baseline(v_pool)[...,idx] + baseline(vg))/3
// ---------------------------------------------------------------------------
__global__ void k_final(const float* __restrict__ Vrel,
                        const float* __restrict__ VG,
                        const float* __restrict__ pool,
                        const int* __restrict__ indices,
                        const int* __restrict__ counts,
                        const float* __restrict__ wt,
                        const float* __restrict__ wc,
                        float* __restrict__ out) {
  int idx = blockIdx.x * blockDim.x + threadIdx.x;
  if (idx >= C_OUT * T_PRED * N_PED) return;
  int o = idx / (T_PRED * N_PED);
  int rem = idx % (T_PRED * N_PED);
  int p = rem / N_PED;
  int n = rem % N_PED;
  int g = indices[n];
  float invc = 1.f / fmaxf((float)counts[g], 1.f);
  float b1 = 0.f, b2 = 0.f, b3 = 0.f;
  for (int c = 0; c < C_IN; ++c) {
    float wco = wc[c * C_OUT + o];
    for (int t = 0; t < T_OBS; ++t) {
      float wtp = wt[t * T_PRED + p] * wco;
      int ct = c * T_OBS + t;
      b1 += Vrel[ct * N_PED + n] * wtp;
      b3 += VG[ct * N_PED + n] * wtp;
      b2 += pool[ct * N_PED + g] * invc * wtp;
    }
  }
  out[idx] = (b1 + b2 + b3) * (1.f / 3.f);
}

// ---------------------------------------------------------------------------
extern "C" void kernel_launch(void* const* d_in, const int* in_sizes, int n_in,
                              void* d_out, int out_size, void* d_ws, size_t ws_size,
                              hipStream_t stream) {
  const float* v_abs = (const float*)d_in[0];
  const float* v_rel = (const float*)d_in[1];
  const float* vel_w = (const float*)d_in[2];
  const float* vel_b = (const float*)d_in[3];
  const float* g1_w  = (const float*)d_in[4];
  const float* g1_b  = (const float*)d_in[5];
  const float* bn_g  = (const float*)d_in[6];
  const float* bn_b  = (const float*)d_in[7];
  const float* g2_w  = (const float*)d_in[8];
  const float* g2_b  = (const float*)d_in[9];
  const float* b_wt  = (const float*)d_in[10];
  const float* b_wc  = (const float*)d_in[11];
  float* out = (float*)d_out;

  // workspace layout (floats)
  float* ws = (float*)d_ws;
  size_t off = 0;
  float* X      = ws + off; off += (size_t)CG_IN * N_PED;       // 24576
  float* Y      = ws + off; off += (size_t)HID * N_PED;         // 12288
  float* DM     = ws + off; off += (size_t)N_PED * N_PED;       // 589824
  float* colsum = ws + off; off += N_PED;                        // 768
  float* VG     = ws + off; off += (size_t)C_IN * T_OBS * N_PED; // 12288
  float* pool   = ws + off; off += (size_t)C_IN * T_OBS * N_PED; // 12288
  int* indices  = (int*)(ws + off); off += N_PED;                // 768
  int* counts   = (int*)(ws + off); off += N_PED;                // 768
  if (ws_size < off * sizeof(float)) return;

  // 1) features X = [v_abs ; conv1x1(v_rel)]
  k_build_X<<<(CG_IN * N_PED + 255) / 256, 256, 0, stream>>>(v_abs, v_rel, vel_w, vel_b, X);

  // 2) Y = W1 @ X  (fp32 WMMA)
  k_y_wmma<<<N_PED / 16, 32, 0, stream>>>(g1_w, X, Y);

  // 3) fused relu/BN/dot/exp/symmetrize -> dist_mean
  k_dist<<<(N_PED * N_PED + 255) / 256, 256, 0, stream>>>(Y, g1_b, bn_g, bn_b, g2_w, g2_b, DM);

  // 4) sigmoid column sums
  k_colsum<<<N_PED, 256, 0, stream>>>(DM, colsum);

  // 5) v_soft GEMM (fp32 WMMA, sigmoid fused) + straight-through v_grouped
  k_vsoft_wmma<<<N_PED / 16, 32, 0, stream>>>(v_rel, DM, colsum, VG);

  // 6) serial union-find relabeling + ranking
  k_group<<<1, N_PED, 0, stream>>>(DM, indices, counts);

  // 7) group pooling
  k_zero<<<(C_IN * T_OBS * N_PED + 255) / 256, 256, 0, stream>>>(pool, C_IN * T_OBS * N_PED);
  k_scatter<<<(N_PED + 255) / 256, 256, 0, stream>>>(VG, indices, pool);

  // 8) three baseline branches + mean
  k_final<<<(C_OUT * T_PRED * N_PED + 255) / 256, 256, 0, stream>>>(
      v_rel, VG, pool, indices, counts, b_wt, b_wc, out);
}